// NonMaximaSuppression3d_13675175870970
// MI455X (gfx1250) — compile-verified
//
#include <hip/hip_runtime.h>
#include <stdint.h>

// Problem shape (fixed by the reference): x = (B=2, C=1, D=128, H=256, W=256) fp32
#define Bt 2
#define Dd 128
#define Hh 256
#define Ww 256

// Tile: each 256-thread block produces TD x TH x TW outputs from an LDS-staged
// halo tile of (TD+2) x (TH+2) x (TW+2). All dims divide the volume evenly.
#define TD 4
#define TH 8
#define TW 32
#define LD (TD + 2)
#define LH (TH + 2)
#define LW (TW + 2)
#define TILE_ELEMS (LD * LH * LW)  // 6*10*34 = 2040 floats = 8160 B LDS

__global__ __launch_bounds__(256) void nms3d_kernel(const float* __restrict__ src,
                                                    float* __restrict__ dst) {
    __shared__ float tile[TILE_ELEMS];

    const int tid = threadIdx.x;
    const int x0 = blockIdx.x * TW;
    const int y0 = blockIdx.y * TH;
    const int zt = blockIdx.z;                 // 0 .. Bt*(Dd/TD)-1
    const int batch = zt / (Dd / TD);
    const int z0 = (zt % (Dd / TD)) * TD;

    // LDS byte offset of the tile: flat LDS pointers keep the offset in the
    // low 32 bits.
    const unsigned lds_base = (unsigned)(uintptr_t)(&tile[0]);

    const size_t batch_off = (size_t)batch * Dd * Hh * Ww;

    // ---- Stage halo tile into LDS via gfx1250 async copy (ASYNCcnt path) ----
    // Replicate padding = clamp global coords per dimension (depth clamped
    // within this batch's [0, Dd-1] range).
#pragma unroll
    for (int it = 0; it < (TILE_ELEMS + 255) / 256; ++it) {
        int i = it * 256 + tid;
        if (i < TILE_ELEMS) {
            int lz = i / (LH * LW);
            int r  = i - lz * (LH * LW);
            int ly = r / LW;
            int lx = r - ly * LW;
            int gz = z0 - 1 + lz; gz = gz < 0 ? 0 : (gz > Dd - 1 ? Dd - 1 : gz);
            int gy = y0 - 1 + ly; gy = gy < 0 ? 0 : (gy > Hh - 1 ? Hh - 1 : gy);
            int gx = x0 - 1 + lx; gx = gx < 0 ? 0 : (gx > Ww - 1 ? Ww - 1 : gx);
            const float* gp = src + batch_off + ((size_t)gz * Hh + gy) * Ww + gx;
            unsigned lds_addr = lds_base + (unsigned)i * 4u;
            uint64_t ga = (uint64_t)(uintptr_t)gp;
            // GLOBAL_LOAD_ASYNC_TO_LDS_B32: each lane copies one dword from global memory to LDS.
            asm volatile("global_load_async_to_lds_b32 %0, %1, off"
                         :: "v"(lds_addr), "v"(ga)
                         : "memory");
        }
    }
    // Drain this wave's async copies, then sync the workgroup so every wave
    // sees the full tile.
    asm volatile("s_wait_asynccnt 0" ::: "memory");
    __syncthreads();

    // ---- Compute: factorized 26-neighbor max ----
    // Per plane p: ring = max of the 8 in-plane neighbors, full = max of all 9.
    // Output at depth dz uses full(dz) (plane below), ring(dz+1) (own plane,
    // center excluded), full(dz+2) (plane above), plus the implicit 0.
    const int tx = tid & (TW - 1);
    const int ty = tid >> 5;

    float ringm[LD], fullm[LD], cent[LD];
#pragma unroll
    for (int p = 0; p < LD; ++p) {
        const float* row0 = &tile[(p * LH + ty    ) * LW + tx];
        const float* row1 = &tile[(p * LH + ty + 1) * LW + tx];
        const float* row2 = &tile[(p * LH + ty + 2) * LW + tx];
        float a0 = row0[0], a1 = row0[1], a2 = row0[2];
        float b0 = row1[0], b1 = row1[1], b2 = row1[2];
        float c0 = row2[0], c1 = row2[1], c2 = row2[2];
        float ring = fmaxf(fmaxf(fmaxf(a0, a1), fmaxf(a2, b0)),
                           fmaxf(fmaxf(b2, c0), fmaxf(c1, c2)));
        ringm[p] = ring;
        fullm[p] = fmaxf(ring, b1);
        cent[p]  = b1;
    }

#pragma unroll
    for (int dz = 0; dz < TD; ++dz) {
        float m = fmaxf(fmaxf(fullm[dz], fullm[dz + 2]), ringm[dz + 1]);
        m = fmaxf(m, 0.0f);  // kornia center-channel-is-zero quirk
        float xv = cent[dz + 1];
        float o  = (xv > m) ? xv : 0.0f;
        size_t oidx = batch_off + ((size_t)(z0 + dz) * Hh + (y0 + ty)) * Ww + (x0 + tx);
        dst[oidx] = o;
    }
}

extern "C" void kernel_launch(void* const* d_in, const int* in_sizes, int n_in,
                              void* d_out, int out_size, void* d_ws, size_t ws_size,
                              hipStream_t stream) {
    (void)in_sizes; (void)n_in; (void)out_size; (void)d_ws; (void)ws_size;
    const float* x = (const float*)d_in[0];
    float* out = (float*)d_out;
    dim3 grid(Ww / TW, Hh / TH, Bt * (Dd / TD));  // (8, 32, 64) = 16384 blocks
    nms3d_kernel<<<grid, dim3(256, 1, 1), 0, stream>>>(x, out);
}